// BernNet_36086315221678
// MI455X (gfx1250) — compile-verified
//
#include <hip/hip_runtime.h>
#include <hip/hip_bf16.h>
#include <stdint.h>
#include <stddef.h>

typedef __attribute__((ext_vector_type(2))) float v2f;
typedef __attribute__((ext_vector_type(8))) float v8f;

__device__ __constant__ float BINOM15[16] = {
    1.f, 15.f, 105.f, 455.f, 1365.f, 3003.f, 5005.f, 6435.f,
    6435.f, 5005.f, 3003.f, 1365.f, 455.f, 105.f, 15.f, 1.f};

// ---------------- small utility kernels ----------------

__global__ void zero_int_kernel(int* p, int n) {
  int i = blockIdx.x * blockDim.x + threadIdx.x;
  if (i < n) p[i] = 0;
}

// count non-self edges per source row (== float degree in the reference)
__global__ void edge_count_kernel(const int* __restrict__ ei, int E,
                                  int* __restrict__ cnt) {
  int e = blockIdx.x * blockDim.x + threadIdx.x;
  if (e >= E) return;
  int r = ei[e];
  int c = ei[E + e];
  if (r != c) atomicAdd(&cnt[r], 1);
}

__global__ void dinv_kernel(const int* __restrict__ cnt, float* __restrict__ dinv, int N) {
  int i = blockIdx.x * blockDim.x + threadIdx.x;
  if (i >= N) return;
  float d = (float)cnt[i];
  dinv[i] = (d > 0.f) ? rsqrtf(d) : 0.f;
}

// single-block exclusive scan over N row counts -> row_ptr[0..N], cursor copy
__global__ void scan_kernel(const int* __restrict__ cnt, int* __restrict__ rowp,
                            int* __restrict__ cursor, int N) {
  __shared__ int s[1024];
  __shared__ int carry;
  int tid = threadIdx.x;
  if (tid == 0) carry = 0;
  __syncthreads();
  for (int base = 0; base < N; base += 1024) {
    int i = base + tid;
    int v = (i < N) ? cnt[i] : 0;
    s[tid] = v;
    __syncthreads();
    for (int off = 1; off < 1024; off <<= 1) {
      int t = (tid >= off) ? s[tid - off] : 0;
      __syncthreads();
      s[tid] += t;
      __syncthreads();
    }
    int cbase = carry;
    int excl = cbase + s[tid] - v;
    if (i < N) { rowp[i] = excl; cursor[i] = excl; }
    __syncthreads();
    if (tid == 0) carry = cbase + s[1023];
    __syncthreads();
  }
  if (tid == 0) rowp[N] = carry;
}

__global__ void edge_scatter_kernel(const int* __restrict__ ei, int E,
                                    const float* __restrict__ dinv,
                                    int* __restrict__ cursor,
                                    int* __restrict__ col_s,
                                    float* __restrict__ ew_s) {
  int e = blockIdx.x * blockDim.x + threadIdx.x;
  if (e >= E) return;
  int r = ei[e];
  int c = ei[E + e];
  if (r != c) {
    int pos = atomicAdd(&cursor[r], 1);
    col_s[pos] = c;
    ew_s[pos] = dinv[r] * dinv[c];
  }
}

// c[m] = relu(temp[m]) * C(15,m) / 2^15 for both temp vectors
__global__ void coef_kernel(const float* __restrict__ t1, const float* __restrict__ t3,
                            float* __restrict__ c1, float* __restrict__ c3) {
  int j = threadIdx.x;
  if (j >= 16) return;
  const float scale = 1.0f / 32768.0f;
  c1[j] = fmaxf(t1[j], 0.f) * BINOM15[j] * scale;
  c3[j] = fmaxf(t3[j], 0.f) * BINOM15[j] * scale;
}

// R = coef[0] * y
__global__ void scale_init_kernel(const float* __restrict__ y,
                                  const float* __restrict__ coef,
                                  float* __restrict__ R, size_t n) {
  size_t i = (size_t)blockIdx.x * blockDim.x + threadIdx.x;
  if (i >= n) return;
  R[i] = coef[0] * y[i];
}

// zero-pad a KxN row-major matrix into KxNpad (Npad multiple of 16)
__global__ void pad_matrix_kernel(const float* __restrict__ src, float* __restrict__ dst,
                                  int K, int N, int Npad) {
  int i = blockIdx.x * blockDim.x + threadIdx.x;
  if (i >= K * Npad) return;
  int r = i / Npad;
  int c = i - r * Npad;
  dst[i] = (c < N) ? src[r * N + c] : 0.f;
}

// ------- WMMA fp32 GEMM: C[M x Nout] = [relu](A[MxK] @ B[K x Npad] + bias) -------
// B is pre-padded to Npad columns (multiple of 16) so the inner K-loop has NO
// per-lane guards: 1x b64 A load + 2x coalesced b32 B loads + v_wmma_f32_16x16x4_f32.
// Nout (<= Npad) is only used for the bias load and the epilogue store guard.
__global__ void gemm_wmma_f32(const float* __restrict__ A, const float* __restrict__ B,
                              const float* __restrict__ bias, float* __restrict__ C,
                              int M, int Nout, int Npad, int K, int doRelu) {
  int wpb = blockDim.x >> 5;
  int wid = blockIdx.x * wpb + ((int)threadIdx.x >> 5);
  int lane = (int)threadIdx.x & 31;
  int ntiles = Npad >> 4;
  int mt = wid / ntiles;
  int nt = wid - mt * ntiles;
  if (mt * 16 >= M) return;  // whole wave exits together; EXEC stays all-1s for WMMA

  int half = lane >> 4;  // 0: K pair {k,k+1} / M 0..7 ; 1: K pair {k+2,k+3} / M 8..15
  int lr = lane & 15;
  int m0 = mt << 4;
  int n0 = nt << 4;
  int colN = n0 + lr;                 // always < Npad (valid for B loads)
  bool colOK = (colN < Nout);         // epilogue-only guard

  float bv = colOK ? bias[colN] : 0.0f;
  v8f acc;
#pragma unroll
  for (int j = 0; j < 8; ++j) acc[j] = bv;

  const float* Ar = A + (size_t)(m0 + lr) * K + 2 * half;  // 16x4 A-tile layout
  const float* Bq = B + (size_t)(2 * half) * Npad + colN;  // 4x16 B-tile layout
  for (int k = 0; k < K; k += 4) {
    v2f a;
    a.x = Ar[0];
    a.y = Ar[1];
    v2f b;
    b.x = Bq[0];
    b.y = Bq[Npad];
    acc = __builtin_amdgcn_wmma_f32_16x16x4_f32(false, a, false, b, (short)0, acc,
                                                false, false);
    Ar += 4;
    Bq += (size_t)4 * Npad;
  }

#pragma unroll
  for (int j = 0; j < 8; ++j) {
    float v = acc[j];
    if (doRelu) v = fmaxf(v, 0.0f);
    int mrow = m0 + j + 8 * half;
    if (colOK) C[(size_t)mrow * Nout + colN] = v;
  }
}

// ---------------- fused CSR SpMM: one wave32 per row ----------------
// acc = adj(y)[row]; mode 0: out = base - acc           (t_m = L t_{m-1})
//                    mode 1: out = base + acc + c*extra (R_m = P R_{m-1} + c_m t_m)
template <int D, int NV>
__global__ void spmm_row_kernel(const int* __restrict__ rp, const int* __restrict__ cols,
                                const float* __restrict__ ews, const float* __restrict__ y,
                                const float* __restrict__ base,
                                const float* __restrict__ extra,
                                const float* __restrict__ coef, int coefIdx, int mode,
                                float* __restrict__ out, int N) {
  int wpb = blockDim.x >> 5;
  int r = blockIdx.x * wpb + ((int)threadIdx.x >> 5);
  int lane = (int)threadIdx.x & 31;
  if (r >= N) return;

  int e0 = rp[r], e1 = rp[r + 1];
  float acc[NV];
#pragma unroll
  for (int k = 0; k < NV; ++k) acc[k] = 0.f;

  for (int e = e0; e < e1; ++e) {
    int c = cols[e];
    float w = ews[e];
    if (e + 1 < e1) {
      // hide gather latency: prefetch next edge's source row (global_prefetch_b8)
      __builtin_prefetch(y + (size_t)cols[e + 1] * D + lane, 0, 3);
    }
    const float* yr = y + (size_t)c * D;
#pragma unroll
    for (int k = 0; k < NV; ++k) {
      int f = lane + 32 * k;
      if (f < D) acc[k] += w * yr[f];
    }
  }

  float cf = (mode == 1) ? coef[coefIdx] : 0.f;
  size_t ro = (size_t)r * D;
#pragma unroll
  for (int k = 0; k < NV; ++k) {
    int f = lane + 32 * k;
    if (f < D) {
      float v;
      if (mode == 0)
        v = base[ro + f] - acc[k];
      else
        v = base[ro + f] + acc[k] + cf * extra[ro + f];
      out[ro + f] = v;
    }
  }
}

// ---------------- host driver ----------------

extern "C" void kernel_launch(void* const* d_in, const int* in_sizes, int n_in,
                              void* d_out, int out_size, void* d_ws, size_t ws_size,
                              hipStream_t stream) {
  const float* x = (const float*)d_in[0];
  const int* ei = (const int*)d_in[1];
  const float* W1 = (const float*)d_in[2];
  const float* b1 = (const float*)d_in[3];
  const float* W2 = (const float*)d_in[4];
  const float* b2 = (const float*)d_in[5];
  const float* temp1 = (const float*)d_in[6];
  const float* temp3 = (const float*)d_in[7];

  const int HID = in_sizes[3];          // 256
  const int CLS = in_sizes[5];          // 40
  const int FEATS = in_sizes[2] / HID;  // 512
  const int N = in_sizes[0] / FEATS;    // 20000
  const int E = in_sizes[1] / 2;        // 320000
  const int K = 15;
  const int CLSP = ((CLS + 15) / 16) * 16;  // 48

  // ---- carve workspace (~86 MB) ----
  char* p = (char*)d_ws;
  auto carve = [&](size_t bytes) -> char* {
    char* r = p;
    p += (bytes + 255) & ~(size_t)255;
    return r;
  };
  float* h1 = (float*)carve((size_t)N * HID * 4);    // GEMM1 out / t ping-pong
  float* tbuf = (float*)carve((size_t)N * HID * 4);  // t ping-pong
  float* R0 = (float*)carve((size_t)N * HID * 4);    // R ping-pong
  float* R1 = (float*)carve((size_t)N * HID * 4);    // R ping-pong
  float* ew_s = (float*)carve((size_t)E * 4);
  float* dinv = (float*)carve((size_t)N * 4);
  float* c1 = (float*)carve(64);
  float* c3 = (float*)carve(64);
  float* W2p = (float*)carve((size_t)HID * CLSP * 4);  // zero-padded W2 (256x48)
  int* cnt = (int*)carve((size_t)N * 4);
  int* rowp = (int*)carve((size_t)(N + 1) * 4);
  int* cursor = (int*)carve((size_t)N * 4);
  int* col_s = (int*)carve((size_t)E * 4);
  if ((size_t)(p - (char*)d_ws) > ws_size) return;  // insufficient scratch

  const int tb = 256;

  // ---- graph preprocessing: degree -> dinv -> CSR(col, ew) ----
  zero_int_kernel<<<(N + tb - 1) / tb, tb, 0, stream>>>(cnt, N);
  edge_count_kernel<<<(E + tb - 1) / tb, tb, 0, stream>>>(ei, E, cnt);
  dinv_kernel<<<(N + tb - 1) / tb, tb, 0, stream>>>(cnt, dinv, N);
  scan_kernel<<<1, 1024, 0, stream>>>(cnt, rowp, cursor, N);
  edge_scatter_kernel<<<(E + tb - 1) / tb, tb, 0, stream>>>(ei, E, dinv, cursor, col_s, ew_s);
  coef_kernel<<<1, 32, 0, stream>>>(temp1, temp3, c1, c3);
  pad_matrix_kernel<<<(HID * CLSP + tb - 1) / tb, tb, 0, stream>>>(W2, W2p, HID, CLS, CLSP);

  // ---- GEMM1: h1 = relu(x @ W1 + b1)   (HID already multiple of 16: no padding) ----
  {
    long waves = (long)((N + 15) / 16) * (HID / 16);
    int blocks = (int)((waves + 7) / 8);
    gemm_wmma_f32<<<blocks, 256, 0, stream>>>(x, W1, b1, h1, N, HID, HID, FEATS, 1);
  }

  // ---- prop1 (D=HID): Horner  t_m = L t_{m-1},  R_m = P R_{m-1} + c_m t_m ----
  float* tcur = h1;
  float* tnext = tbuf;
  float* Rcur = R0;
  float* Rnext = R1;
  {
    size_t n = (size_t)N * HID;
    scale_init_kernel<<<(int)((n + 255) / 256), 256, 0, stream>>>(h1, c1, Rcur, n);
    int blocks = (N + 7) / 8;  // 8 waves (rows) per 256-thread block
    for (int m = 1; m <= K; ++m) {
      spmm_row_kernel<256, 8><<<blocks, 256, 0, stream>>>(rowp, col_s, ew_s, tcur, tcur,
                                                          tcur, c1, 0, 0, tnext, N);
      spmm_row_kernel<256, 8><<<blocks, 256, 0, stream>>>(rowp, col_s, ew_s, Rcur, Rcur,
                                                          tnext, c1, m, 1, Rnext, N);
      float* t_ = tcur; tcur = tnext; tnext = t_;
      t_ = Rcur; Rcur = Rnext; Rnext = t_;
    }
  }
  float* hprop = Rcur;  // [N, HID] in R0 or R1; h1/tbuf region now free

  // ---- GEMM2: g2 = hprop @ W2 + b2 ----  (prop2 buffers reuse the free h1 region)
  float* g2 = h1;
  float* t2 = g2 + (size_t)N * CLS;
  float* R0b = t2 + (size_t)N * CLS;
  float* R1b = R0b + (size_t)N * CLS;
  {
    long waves = (long)((N + 15) / 16) * (CLSP / 16);
    int blocks = (int)((waves + 7) / 8);
    gemm_wmma_f32<<<blocks, 256, 0, stream>>>(hprop, W2p, b2, g2, N, CLS, CLSP, HID, 0);
  }

  // ---- prop2 (D=CLS): final P-step writes d_out directly ----
  {
    float* outp = (float*)d_out;
    size_t n = (size_t)N * CLS;
    float* tc = g2; float* tn = t2;
    float* Rc = R0b; float* Rn = R1b;
    scale_init_kernel<<<(int)((n + 255) / 256), 256, 0, stream>>>(g2, c3, Rc, n);
    int blocks = (N + 7) / 8;
    for (int m = 1; m <= K; ++m) {
      spmm_row_kernel<40, 2><<<blocks, 256, 0, stream>>>(rowp, col_s, ew_s, tc, tc, tc,
                                                         c3, 0, 0, tn, N);
      float* Rdst = (m == K) ? outp : Rn;
      spmm_row_kernel<40, 2><<<blocks, 256, 0, stream>>>(rowp, col_s, ew_s, Rc, Rc, tn,
                                                         c3, m, 1, Rdst, N);
      float* t_ = tc; tc = tn; tn = t_;
      t_ = Rc; Rc = Rn; Rn = t_;
    }
  }
}